// MVCTNetSetAbstraction_8211977470434
// MI455X (gfx1250) — compile-verified
//
#include <hip/hip_runtime.h>
#include <hip/hip_bf16.h>

// ---------------- CDNA5 WMMA types ----------------
typedef float v2f __attribute__((ext_vector_type(2)));
typedef float v8f __attribute__((ext_vector_type(8)));

#define WMMA_F32_16x16x4(a, bv, c) \
    __builtin_amdgcn_wmma_f32_16x16x4_f32(false, (a), false, (bv), (short)0, (c), false, false)

// problem constants (from setup_inputs)
constexpr int BB = 4;
constexpr int NN = 8192;
constexpr int SS = 2048;
constexpr int KK = 32;
constexpr float FEPS = 1e-7f;

// ---------------- LDS addressing for async copies ----------------
typedef __attribute__((address_space(3))) const void* lds_cptr_t;

// Issue one GLOBAL_LOAD_ASYNC_TO_LDS_B128: copies 16 bytes per lane from global gsrc to LDS lds_dst.
// Tracked by ASYNCcnt; completion enforced with s_wait_asynccnt + barrier.
__device__ inline void async_b128_to_lds(const float4* gsrc, const float4* lds_dst) {
    unsigned ldsoff = (unsigned)(size_t)(lds_cptr_t)(const void*)lds_dst;
    unsigned long long ga = (unsigned long long)(size_t)gsrc;
    asm volatile("global_load_async_to_lds_b128 %0, %1, off"
                 :: "v"(ldsoff), "v"(ga) : "memory");
}

__device__ inline void wait_async0() {
    asm volatile("s_wait_asynccnt 0x0" ::: "memory");
}

// ---------------- wave32 helpers ----------------
__device__ inline float wave_max_f(float v) {
#pragma unroll
    for (int m = 16; m; m >>= 1) v = fmaxf(v, __shfl_xor(v, m));
    return v;
}

// lane index of max value; ties -> lowest lane. All lanes converge to same answer.
__device__ inline int wave_argmax(float v, int lane) {
    int l = lane;
#pragma unroll
    for (int m = 16; m; m >>= 1) {
        float ov = __shfl_xor(v, m);
        int ol = __shfl_xor(l, m);
        if (ov > v || (ov == v && ol < l)) { v = ov; l = ol; }
    }
    return l;
}

// full-wave bitonic sort ascending by key, payload follows
__device__ inline void bitonic32_asc(float& key, int& pay, int lane) {
#pragma unroll
    for (int k = 2; k <= 32; k <<= 1) {
#pragma unroll
        for (int j = k >> 1; j > 0; j >>= 1) {
            float ok = __shfl_xor(key, j);
            int op = __shfl_xor(pay, j);
            bool up = ((lane & k) == 0);
            bool lower = ((lane & j) == 0);
            bool takeMin = (lower == up);
            bool doSwap = takeMin ? (ok < key) : (ok > key);
            if (doSwap) { key = ok; pay = op; }
        }
    }
}

// ---------------- stage A: pack points as (x,y,z,|p|^2) ----------------
__global__ void pack_kernel(const float* __restrict__ xyz, float4* __restrict__ P4) {
    int t = blockIdx.x * 256 + threadIdx.x;
    if (t >= BB * NN) return;
    float x = xyz[3 * t + 0], y = xyz[3 * t + 1], z = xyz[3 * t + 2];
    P4[t] = make_float4(x, y, z, x * x + y * y + z * z);
}

// gather query points / normals
__global__ void gatherq_kernel(const float* __restrict__ xyz, const float* __restrict__ nrm,
                               const int* __restrict__ sidx, float4* __restrict__ Q4,
                               float4* __restrict__ Qn4) {
    int t = blockIdx.x * 256 + threadIdx.x;
    if (t >= BB * SS) return;
    int b = t / SS;
    int i = sidx[t];
    const float* p = xyz + (size_t)(b * NN + i) * 3;
    const float* n = nrm + (size_t)(b * NN + i) * 3;
    Q4[t] = make_float4(p[0], p[1], p[2], 0.f);
    Qn4[t] = make_float4(n[0], n[1], n[2], 0.f);
}

// ---------------- stage B: KNN via V_WMMA_F32_16X16X4_F32 ----------------
// score(m,n) = -2 q_m . p_n + |p_n|^2   (|q|^2 dropped: constant per row)
// A row m = (-2qx, -2qy, -2qz, 1);  B col n = (px, py, pz, |p|^2)
// Point panels are double-buffered in LDS via GLOBAL_LOAD_ASYNC_TO_LDS_B128 so the
// global-memory latency hides behind the previous panel's top-k selection phase.
__global__ __launch_bounds__(512) void knn_kernel(const float4* __restrict__ P4,
                                                  const float4* __restrict__ Q4,
                                                  int* __restrict__ knn) {
    __shared__ float sc[16][260];       // 16 queries x 256-col score panel (+pad)
    __shared__ float4 pts[2][256];      // double-buffered point panel (4KB each)
    const int lane = threadIdx.x & 31;
    const int w = threadIdx.x >> 5;  // wave id 0..15
    const int b = blockIdx.x >> 7;   // SS/16 = 128 tiles per batch
    const int tile = blockIdx.x & 127;
    const int qbase = tile * 16;
    const bool lo = (lane < 16);
    const int m16 = lane & 15;
    constexpr int NPANEL = NN / 256;

    // A matrix (same 16 queries for all waves in WG)
    float4 q = Q4[b * SS + qbase + m16];
    v2f A;
    A.x = lo ? -2.f * q.x : -2.f * q.z;  // K = 0 / 2
    A.y = lo ? -2.f * q.y : 1.0f;        // K = 1 / 3
    const float4* P4b = P4 + (size_t)b * NN;

    // running top-32: one candidate per lane
    float cv = 3.4e38f;
    int ci = 0;
    float kmax = 3.4e38f;

    // prologue: async-stage panel 0 (waves 0-7, one B128 per lane)
    if (threadIdx.x < 256) async_b128_to_lds(P4b + threadIdx.x, &pts[0][threadIdx.x]);

    int buf = 0;
    for (int panel = 0; panel < NPANEL; ++panel) {
        wait_async0();       // issuing waves: panel's copy done (no-op for waves 8-15)
        __syncthreads();     // pts[buf] visible to all; sc free for reuse
        if (panel + 1 < NPANEL && threadIdx.x < 256)
            async_b128_to_lds(P4b + (panel + 1) * 256 + threadIdx.x,
                              &pts[buf ^ 1][threadIdx.x]);

        // --- compute phase: wave w fills columns [w*16, w*16+16) of the panel
        const float2* pp = (const float2*)&pts[buf][0];
        float2 pb = pp[(w * 16 + m16) * 2 + (lo ? 0 : 1)];  // (x,y) or (z,|p|^2)
        v2f Bv;
        Bv.x = pb.x;  // K = 0 / 2
        Bv.y = pb.y;  // K = 1 / 3
        v8f d = {};
        d = WMMA_F32_16x16x4(A, Bv, d);
        int colbase = w * 16 + m16;
        int rbase = lo ? 0 : 8;
#pragma unroll
        for (int i = 0; i < 8; ++i) sc[rbase + i][colbase] = d[i];
        __syncthreads();

        // --- select phase: wave w merges row w (its query) into running top-32
#pragma unroll 1
        for (int t = 0; t < 8; ++t) {
            int col = t * 32 + lane;
            float v = sc[w][col];
            int id = panel * 256 + col;
            unsigned long long m = __ballot(v < kmax);
            while (m) {
                int src = __ffsll(m) - 1;
                m &= m - 1;
                float nv = __shfl(v, src);
                int nid = __shfl(id, src);
                if (nv < kmax) {  // wave-uniform
                    int vic = wave_argmax(cv, lane);
                    if (lane == vic) { cv = nv; ci = nid; }
                    kmax = wave_max_f(cv);
                }
            }
        }
        buf ^= 1;
    }
    // order by distance ascending (mimics top_k output order)
    bitonic32_asc(cv, ci, lane);
    knn[(size_t)(b * SS + qbase + w) * KK + lane] = ci;
}

// ---------------- stage C: ordering + RISP features + MLP(14->64) + max_k ----------------
__global__ __launch_bounds__(256) void feat_mlp_kernel(const float* __restrict__ xyz,
                                                       const float* __restrict__ nrm,
                                                       const float4* __restrict__ Q4,
                                                       const float4* __restrict__ Qn4,
                                                       const int* __restrict__ knn,
                                                       const float* __restrict__ W,
                                                       const float* __restrict__ bias,
                                                       float* __restrict__ out) {
    __shared__ float F[8][32][17];   // per-wave 32(k) x 16(feat, padded) matrix
    __shared__ float Wsh[16 * 64];   // W zero-padded 14 -> 16 rows
    const int lane = threadIdx.x & 31;
    const int w = threadIdx.x >> 5;
    const int qs = blockIdx.x * 8 + w;  // flat (b*S + s)
    const int b = qs / SS;

    __builtin_prefetch(W, 0, 0);  // global_prefetch_b8 of weights

    // stage W into LDS (zero-padded rows 14,15)
#pragma unroll
    for (int i = threadIdx.x; i < 16 * 64; i += 256) {
        int row = i >> 6;
        Wsh[i] = (row < 14) ? W[i] : 0.f;
    }

    // lane = k-th nearest neighbor (distance-sorted from knn kernel)
    int ni = knn[(size_t)qs * KK + lane];
    const float* p = xyz + (size_t)(b * NN + ni) * 3;
    const float* pn = nrm + (size_t)(b * NN + ni) * 3;
    float4 q = Q4[qs], qn = Qn4[qs];

    float xix = p[0] - q.x, xiy = p[1] - q.y, xiz = p[2] - q.z;
    float nx = pn[0], ny = pn[1], nz = pn[2];

    // ---- order_index: project to tangent plane, sort by signed angle dot vs ref ----
    float dp = xix * qn.x + xiy * qn.y + xiz * qn.z;
    float px = xix - dp * qn.x, py = xiy - dp * qn.y, pz = xiz - dp * qn.z;
    float plen = sqrtf(px * px + py * py + pz * pz);
    float inv = (plen > 0.f) ? (1.f / plen) : 0.f;
    float pux = px * inv, puy = py * inv, puz = pz * inv;
    int rl = wave_argmax(plen, lane);
    float rx = __shfl(pux, rl), ry = __shfl(puy, rl), rz = __shfl(puz, rl);
    float dots = pux * rx + puy * ry + puz * rz;
    float cxv = puy * rz - puz * ry;
    float cyv = puz * rx - pux * rz;
    float czv = pux * ry - puy * rx;
    float sg = cxv * qn.x + cyv * qn.y + czv * qn.z;
    float sgn = (sg > 0.f) ? 1.f : ((sg < 0.f) ? -1.f : 0.f);
    if (lane == 0) sgn = 1.f;
    dots = sgn * dots - (1.f - sgn);

    float key = -dots;  // argsort(-dots) ascending == dots descending
    int pay = lane;
    bitonic32_asc(key, pay, lane);

    // gather ordered xi / xi_norm
    float ax = __shfl(xix, pay), ay = __shfl(xiy, pay), az = __shfl(xiz, pay);
    float anx = __shfl(nx, pay), any_ = __shfl(ny, pay), anz = __shfl(nz, pay);
    // rolls (num_shifts = 2 since S >= 1024): x3 = roll(+2), x4 = roll(-2)
    int l3 = (lane + 30) & 31, l4 = (lane + 2) & 31;
    float bx = __shfl(ax, l3), by = __shfl(ay, l3), bz = __shfl(az, l3);
    float bnx = __shfl(anx, l3), bny = __shfl(any_, l3), bnz = __shfl(anz, l3);
    float cx = __shfl(ax, l4), cy = __shfl(ay, l4), cz = __shfl(az, l4);
    float cnx = __shfl(anx, l4), cny = __shfl(any_, l4), cnz = __shfl(anz, l4);

    // ---- 14 RISP features (per lane = ordered neighbor k) ----
    float Li = sqrtf(ax * ax + ay * ay + az * az);
    float L3 = sqrtf(bx * bx + by * by + bz * bz);
    float iLi = 1.f / (Li + FEPS), iL3 = 1.f / (L3 + FEPS);
    float f0 = Li;
    float f1 = (ax * qn.x + ay * qn.y + az * qn.z) * iLi;
    float f2 = -(ax * anx + ay * any_ + az * anz) * iLi;
    float f3 = (ax * bx + ay * by + az * bz) * iLi * iL3;
    float f4 = (bx * qn.x + by * qn.y + bz * qn.z) * iL3;
    float f5 = -(bx * bnx + by * bny + bz * bnz) * iL3;
    float ex = bx - ax, ey = by - ay, ez = bz - az;  // x3 - xi
    float L23 = sqrtf(ex * ex + ey * ey + ez * ez);
    float iL23 = 1.f / (L23 + FEPS);
    float f6 = (bx * ex + by * ey + bz * ez) * iL3 * iL23;
    float f7 = (ex * anx + ey * any_ + ez * anz) * iL23;
    float f8 = -(ex * bnx + ey * bny + ez * bnz) * iL23;
    // new_surface_feature(x4, x4n, 0, qn, xi, xin, x3, x3n)
    float L4 = sqrtf(cx * cx + cy * cy + cz * cz);
    float iL4 = 1.f / (L4 + FEPS);
    float pxix = cx * iL4, pxiy = cy * iL4, pxiz = cz * iL4;       // unit(x4)
    float px2x = ax * iLi, px2y = ay * iLi, px2z = az * iLi;       // unit(xi)
    float px3x = bx * iL3, px3y = by * iL3, px3z = bz * iL3;       // unit(x3)
    float gx = cx - ax, gy = cy - ay, gz = cz - az;                // x4 - xi
    float L24 = sqrtf(gx * gx + gy * gy + gz * gz);
    float iL24 = 1.f / (L24 + FEPS);
    float x2x = gx * iL24, x2y = gy * iL24, x2z = gz * iL24;
    float n1x = pxiy * px2z - pxiz * px2y;
    float n1y = pxiz * px2x - pxix * px2z;
    float n1z = pxix * px2y - pxiy * px2x;
    float n2x = px3y * px2z - px3z * px2y;
    float n2y = px3z * px2x - px3x * px2z;
    float n2z = px3x * px2y - px3y * px2x;
    float f9 = pxix * px2x + pxiy * px2y + pxiz * px2z;   // a11
    float f10 = pxix * x2x + pxiy * x2y + pxiz * x2z;     // a12
    float f11 = x2x * cnx + x2y * cny + x2z * cnz;        // a21
    float f12 = px2x * cnx + px2y * cny + px2z * cnz;     // a22
    float f13 = n1x * n2x + n1y * n2y + n1z * n2z;        // a3

    float* Fr = &F[w][lane][0];
    Fr[0] = f0;  Fr[1] = f1;  Fr[2] = f2;  Fr[3] = f3;  Fr[4] = f4;
    Fr[5] = f5;  Fr[6] = f6;  Fr[7] = f7;  Fr[8] = f8;  Fr[9] = f9;
    Fr[10] = f10; Fr[11] = f11; Fr[12] = f12; Fr[13] = f13;
    Fr[14] = 0.f; Fr[15] = 0.f;  // K-pad 14 -> 16
    __syncthreads();  // F + Wsh ready

    // ---- MLP: (32 x 14) @ W(14 x 64) + b, relu, max over k ----
    const bool lo = (lane < 16);
    const int mrow = lane & 15;

    // hoist A fragments: depend only on K-chunk c, reused across all 4 output tiles
    v2f A0[4], A1[4];
#pragma unroll
    for (int c = 0; c < 4; ++c) {
        int k0 = 4 * c + (lo ? 0 : 2);
        int k1 = k0 + 1;
        A0[c].x = F[w][mrow][k0];      A0[c].y = F[w][mrow][k1];
        A1[c].x = F[w][16 + mrow][k0]; A1[c].y = F[w][16 + mrow][k1];
    }

#pragma unroll
    for (int t = 0; t < 4; ++t) {  // output column tiles of 16
        int colN = t * 16 + mrow;
        float bv0 = bias[colN];
        v8f acc0 = {bv0, bv0, bv0, bv0, bv0, bv0, bv0, bv0};
        v8f acc1 = acc0;
#pragma unroll
        for (int c = 0; c < 4; ++c) {  // K chunks of 4 (14 padded to 16)
            int k0 = 4 * c + (lo ? 0 : 2);
            v2f Bv;
            Bv.x = Wsh[k0 * 64 + colN];
            Bv.y = Wsh[(k0 + 1) * 64 + colN];
            acc0 = WMMA_F32_16x16x4(A0[c], Bv, acc0);
            acc1 = WMMA_F32_16x16x4(A1[c], Bv, acc1);
        }
        float m = acc0[0];
#pragma unroll
        for (int i = 1; i < 8; ++i) m = fmaxf(m, acc0[i]);
#pragma unroll
        for (int i = 0; i < 8; ++i) m = fmaxf(m, acc1[i]);
        m = fmaxf(m, 0.f);                 // relu commutes with max
        m = fmaxf(m, __shfl_xor(m, 16));   // combine halves (k rows 0-7 vs 8-15)
        if (lane < 16) out[(size_t)qs * 64 + colN] = m;
    }
}

// ---------------- host launch ----------------
extern "C" void kernel_launch(void* const* d_in, const int* in_sizes, int n_in,
                              void* d_out, int out_size, void* d_ws, size_t ws_size,
                              hipStream_t stream) {
    const float* xyz = (const float*)d_in[0];
    const float* nrm = (const float*)d_in[1];
    const float* W = (const float*)d_in[2];
    const float* bias = (const float*)d_in[3];
    const int* sidx = (const int*)d_in[4];
    float* out = (float*)d_out;

    char* ws = (char*)d_ws;
    size_t o = 0;
    float4* P4 = (float4*)(ws + o); o += (size_t)BB * NN * sizeof(float4);   // 512 KB
    float4* Q4 = (float4*)(ws + o); o += (size_t)BB * SS * sizeof(float4);   // 128 KB
    float4* Qn4 = (float4*)(ws + o); o += (size_t)BB * SS * sizeof(float4);  // 128 KB
    int* knn = (int*)(ws + o);                                               // 1 MB

    pack_kernel<<<(BB * NN + 255) / 256, 256, 0, stream>>>(xyz, P4);
    gatherq_kernel<<<(BB * SS + 255) / 256, 256, 0, stream>>>(xyz, nrm, sidx, Q4, Qn4);
    knn_kernel<<<BB * (SS / 16), 512, 0, stream>>>(P4, Q4, knn);
    feat_mlp_kernel<<<BB * SS / 8, 256, 0, stream>>>(xyz, nrm, Q4, Qn4, knn, W, bias, out);
}